// S4R_88596585382814
// MI455X (gfx1250) — compile-verified
//
#include <hip/hip_runtime.h>
#include <hip/hip_bf16.h>

typedef __attribute__((ext_vector_type(2))) float v2f;
typedef __attribute__((ext_vector_type(8))) float v8f;

#define HCH   1024      // H
#define NMOD  1024      // N
#define LSEQ  4096      // L
#define NF    8192      // 2L FFT size
#define BSZ   8
#define PI_F  3.14159265358979323846f

// ---------------------------------------------------------------------------
// Kernel 1: Amat[h,n] = C[h,n] * B[n]
// ---------------------------------------------------------------------------
__global__ void __launch_bounds__(256) amat_kernel(const float* __restrict__ C,
                                                   const float* __restrict__ Bv,
                                                   float* __restrict__ Amat) {
    int idx = blockIdx.x * 256 + threadIdx.x;
    if (idx < HCH * NMOD) {
        int n = idx & (NMOD - 1);
        Amat[idx] = C[idx] * Bv[n];
    }
}

// ---------------------------------------------------------------------------
// Kernel 2: closed-form spectrum of the truncated mode response in
// BIT-REVERSED frequency order (matches DIF-forward / DIT-inverse FFT).
// G(f) = 0.5*[1/(1-lam z) + 1/(1-conj(lam) z)], z=e^{-i2pi f/NF};
// lam^L underflows to exactly 0 in f32 so the geometric tail vanishes.
// ---------------------------------------------------------------------------
__global__ void __launch_bounds__(256) g_table_kernel(const float* __restrict__ log_decay,
                                                      const float* __restrict__ phase,
                                                      float* __restrict__ G) {
    int idx = blockIdx.x * 256 + threadIdx.x;          // n*NF + j
    if (idx >= NMOD * NF) return;
    int n = idx >> 13;
    int j = idx & (NF - 1);
    unsigned f = __brev((unsigned)j) >> 19;            // 13-bit bit reversal

    float r  = __expf(log_decay[n]);
    float er = __expf(-r);
    float sp, cp; __sincosf(phase[n], &sp, &cp);
    float lr = er * cp, li = er * sp;                  // lambda

    float w = (2.0f * PI_F / (float)NF) * (float)f;
    float zs, zc; __sincosf(w, &zs, &zc);
    float zr = zc, zi = -zs;                           // z = e^{-iw}

    float p1r = 1.0f - (lr * zr - li * zi);
    float p1i =      -(lr * zi + li * zr);
    float i1  = 1.0f / (p1r * p1r + p1i * p1i);
    float t1r = p1r * i1, t1i = -p1i * i1;
    float p2r = 1.0f - (lr * zr + li * zi);
    float p2i =      -(lr * zi - li * zr);
    float i2  = 1.0f / (p2r * p2r + p2i * p2i);
    float t2r = p2r * i2, t2i = -p2i * i2;

    G[(size_t)n * (2 * NF) + j]      = 0.5f * (t1r + t2r);
    G[(size_t)n * (2 * NF) + NF + j] = 0.5f * (t1i + t2i);
}

// ---------------------------------------------------------------------------
// Async staging of a 64-row x 64-col f32 panel into LDS (pitch 68 floats,
// avoids ds bank conflicts on fragment reads). 256 threads x 4 b128 each.
// ---------------------------------------------------------------------------
__device__ __forceinline__ void stage_async(const float* __restrict__ base,
                                            int rowStride, int k0, int c0,
                                            float* buf, int tid) {
#pragma unroll
    for (int i = 0; i < 4; ++i) {
        int s   = tid + 256 * i;
        int row = s >> 4;
        int col = (s & 15) << 2;
        const float* gp = base + (size_t)(k0 + row) * rowStride + c0 + col;
        unsigned lds = (unsigned)(uintptr_t)(buf + row * 68 + col);
        asm volatile("global_load_async_to_lds_b128 %0, %1, off"
                     :: "v"(lds), "v"((unsigned long long)(uintptr_t)gp)
                     : "memory");
    }
}

__device__ __forceinline__ void wait_async_barrier() {
    asm volatile("s_wait_asynccnt 0x0" ::: "memory");
    __syncthreads();
}

// ---------------------------------------------------------------------------
// Kernel 3: Kf = Amat @ G via V_WMMA_F32_16X16X4_F32, cooperative tiling.
// Workgroup = 8 waves: 128 rows x 64 cols. B panel (64x64) double-buffered
// through LDS with async global->LDS copies; each wave: 4 accum tiles.
// ---------------------------------------------------------------------------
__global__ void __launch_bounds__(256) gemm_coop_kernel(const float* __restrict__ A,
                                                        const float* __restrict__ Bm,
                                                        float* __restrict__ Dm,
                                                        int K, int Ncols) {
    __shared__ float stile[2][64 * 68];
    int tid  = threadIdx.x;
    int lane = tid & 31;
    int w    = tid >> 5;
    int row0 = blockIdx.x * 128 + w * 16;
    int c0   = blockIdx.y * 64;
    int lm   = lane & 15;
    int khalf = (lane >> 4) << 1;

    const float* Arow = A + (size_t)(row0 + lm) * K + khalf;

    v8f acc[4];
#pragma unroll
    for (int ct = 0; ct < 4; ++ct) acc[ct] = v8f{0.f,0.f,0.f,0.f,0.f,0.f,0.f,0.f};

    stage_async(Bm, Ncols, 0, c0, &stile[0][0], tid);
    const int nchunks = K >> 6;
    for (int c = 0; c < nchunks; ++c) {
        wait_async_barrier();
        if (c + 1 < nchunks)
            stage_async(Bm, Ncols, (c + 1) << 6, c0, &stile[(c + 1) & 1][0], tid);
        const float* buf = &stile[c & 1][0];
        int kc = c << 6;
        __builtin_prefetch(Arow + kc + 64, 0, 3);
#pragma unroll 4
        for (int k4 = 0; k4 < 16; ++k4) {
            int kk = kc + (k4 << 2);
            v2f af; af.x = Arow[kk]; af.y = Arow[kk + 1];
            const float* bp = buf + ((k4 << 2) + khalf) * 68 + lm;
#pragma unroll
            for (int ct = 0; ct < 4; ++ct) {
                v2f bf; bf.x = bp[ct * 16]; bf.y = bp[ct * 16 + 68];
                acc[ct] = __builtin_amdgcn_wmma_f32_16x16x4_f32(false, af, false, bf,
                                                                (short)0, acc[ct], false, false);
            }
        }
        __syncthreads();
    }
    int mbase = (lane >> 4) ? 8 : 0;
#pragma unroll
    for (int ct = 0; ct < 4; ++ct) {
        float* Dp = Dm + (size_t)row0 * Ncols + c0 + ct * 16 + lm;
#pragma unroll
        for (int r = 0; r < 8; ++r)
            Dp[(size_t)(mbase + r) * Ncols] = acc[ct][r];
    }
}

// ---------------------------------------------------------------------------
// Kernel 4: per-(b,h) row FFT convolution in LDS.
// Twiddles precomputed in a per-stage concatenated LDS table (conflict-free).
// DIF forward (bit-rev out) -> pointwise * Kf(bit-rev) -> DIT inverse.
// ---------------------------------------------------------------------------
__global__ void __launch_bounds__(256) fftconv_kernel(const float* __restrict__ u,
                                                      const float* __restrict__ Kf,
                                                      const float* __restrict__ Dvec,
                                                      float* __restrict__ y) {
    __shared__ float sre[NF];
    __shared__ float sim[NF];
    __shared__ float twr[NF - 1];          // 8191: per-stage tables concatenated
    __shared__ float twi[NF - 1];
    int row = blockIdx.x;                  // b*H + h
    int h   = row & (HCH - 1);
    int tid = threadIdx.x;
    const float* up = u + (size_t)row * LSEQ;

    // fill twiddle tables: stage half-span m at offset (NF - 2m), entries j<m,
    // w = exp(-i*pi*j/m)
    {
        int off = 0;
        for (int m = NF >> 1; m >= 1; m >>= 1) {
            for (int j = tid; j < m; j += 256) {
                float s, c; __sincosf(-PI_F * (float)j / (float)m, &s, &c);
                twr[off + j] = c; twi[off + j] = s;
            }
            off += m;
        }
    }
    for (int i = tid; i < LSEQ; i += 256) { sre[i] = up[i]; sim[i] = 0.f; }
    for (int i = LSEQ + tid; i < NF; i += 256) { sre[i] = 0.f; sim[i] = 0.f; }

    // forward radix-2 DIF
    {
        int off = 0;
        for (int m = NF >> 1; m >= 1; m >>= 1) {
            __syncthreads();
            for (int k = tid; k < (NF >> 1); k += 256) {
                int j  = k & (m - 1);
                int i0 = ((k & ~(m - 1)) << 1) + j;
                int i1 = i0 + m;
                float c = twr[off + j], s = twi[off + j];
                float ar = sre[i0], ai = sim[i0];
                float br = sre[i1], bi = sim[i1];
                float dr = ar - br, di = ai - bi;
                sre[i0] = ar + br; sim[i0] = ai + bi;
                sre[i1] = dr * c - di * s;
                sim[i1] = dr * s + di * c;
            }
            off += m;
        }
    }
    __syncthreads();

    // pointwise spectrum multiply (both in bit-reversed order)
    const float* kfr = Kf + (size_t)h * (2 * NF);
    const float* kfi = kfr + NF;
    for (int j = tid; j < NF; j += 256) {
        float xr = sre[j], xi = sim[j];
        float kr = kfr[j], ki = kfi[j];
        sre[j] = xr * kr - xi * ki;
        sim[j] = xr * ki + xi * kr;
    }

    // inverse radix-2 DIT (conjugated twiddles)
    for (int m = 1; m <= (NF >> 1); m <<= 1) {
        int off = NF - (m << 1);
        __syncthreads();
        for (int k = tid; k < (NF >> 1); k += 256) {
            int j  = k & (m - 1);
            int i0 = ((k & ~(m - 1)) << 1) + j;
            int i1 = i0 + m;
            float c = twr[off + j], s = -twi[off + j];
            float br = sre[i1], bi = sim[i1];
            float tr = br * c - bi * s;
            float ti = br * s + bi * c;
            float ar = sre[i0], ai = sim[i0];
            sre[i0] = ar + tr; sim[i0] = ai + ti;
            sre[i1] = ar - tr; sim[i1] = ai - ti;
        }
    }
    __syncthreads();

    float dh = Dvec[h];
    float* yp = y + (size_t)row * LSEQ;
    const float scale = 1.0f / (float)NF;
    for (int i = tid; i < LSEQ; i += 256)
        yp[i] = sre[i] * scale + dh * up[i];
}

// ---------------------------------------------------------------------------
// Kernel 5: z = W_mix @ y + b_mix ; out = z[:H]*sigmoid(z[H:]) fused.
// Cooperative: workgroup = 128 a-rows (+ paired g-rows) x 64 cols; Y panel
// async-staged through LDS; each wave keeps 4 a-tiles + 4 g-tiles sharing
// the B fragments.
// ---------------------------------------------------------------------------
__global__ void __launch_bounds__(256) mix_gate_kernel(const float* __restrict__ W,
                                                       const float* __restrict__ bmix,
                                                       const float* __restrict__ y,
                                                       float* __restrict__ out) {
    __shared__ float stile[2][64 * 68];
    int tid  = threadIdx.x;
    int lane = tid & 31;
    int w    = tid >> 5;
    int o0   = blockIdx.x * 128 + w * 16;
    int gcol = blockIdx.y * 64;
    int b    = gcol >> 12;                 // / LSEQ
    int l0   = gcol & (LSEQ - 1);
    int lm   = lane & 15;
    int khalf = (lane >> 4) << 1;

    const float* Wa = W + (size_t)(o0 + lm) * HCH + khalf;
    const float* Wg = W + (size_t)(HCH + o0 + lm) * HCH + khalf;
    const float* Yb = y + (size_t)b * HCH * LSEQ;

    v8f acca[4], accg[4];
#pragma unroll
    for (int ct = 0; ct < 4; ++ct) {
        acca[ct] = v8f{0.f,0.f,0.f,0.f,0.f,0.f,0.f,0.f};
        accg[ct] = v8f{0.f,0.f,0.f,0.f,0.f,0.f,0.f,0.f};
    }

    stage_async(Yb, LSEQ, 0, l0, &stile[0][0], tid);
    const int nchunks = HCH >> 6;          // 16
    for (int c = 0; c < nchunks; ++c) {
        wait_async_barrier();
        if (c + 1 < nchunks)
            stage_async(Yb, LSEQ, (c + 1) << 6, l0, &stile[(c + 1) & 1][0], tid);
        const float* buf = &stile[c & 1][0];
        int kc = c << 6;
        __builtin_prefetch(Wa + kc + 64, 0, 3);
        __builtin_prefetch(Wg + kc + 64, 0, 3);
#pragma unroll 2
        for (int k4 = 0; k4 < 16; ++k4) {
            int kk = kc + (k4 << 2);
            v2f af; af.x = Wa[kk]; af.y = Wa[kk + 1];
            v2f gf; gf.x = Wg[kk]; gf.y = Wg[kk + 1];
            const float* bp = buf + ((k4 << 2) + khalf) * 68 + lm;
#pragma unroll
            for (int ct = 0; ct < 4; ++ct) {
                v2f bf; bf.x = bp[ct * 16]; bf.y = bp[ct * 16 + 68];
                acca[ct] = __builtin_amdgcn_wmma_f32_16x16x4_f32(false, af, false, bf,
                                                                 (short)0, acca[ct], false, false);
                accg[ct] = __builtin_amdgcn_wmma_f32_16x16x4_f32(false, gf, false, bf,
                                                                 (short)0, accg[ct], false, false);
            }
        }
        __syncthreads();
    }

    int mbase = (lane >> 4) ? 8 : 0;
#pragma unroll
    for (int ct = 0; ct < 4; ++ct) {
        float* op = out + ((size_t)b * HCH + o0) * LSEQ + l0 + ct * 16 + lm;
#pragma unroll
        for (int r = 0; r < 8; ++r) {
            int m = mbase + r;
            float a = acca[ct][r] + bmix[o0 + m];
            float g = accg[ct][r] + bmix[HCH + o0 + m];
            op[(size_t)m * LSEQ] = a * (1.0f / (1.0f + __expf(-g)));
        }
    }
}

// ---------------------------------------------------------------------------
extern "C" void kernel_launch(void* const* d_in, const int* in_sizes, int n_in,
                              void* d_out, int out_size, void* d_ws, size_t ws_size,
                              hipStream_t stream) {
    const float* u         = (const float*)d_in[0];
    const float* log_decay = (const float*)d_in[1];
    const float* phase     = (const float*)d_in[2];
    const float* C         = (const float*)d_in[3];
    const float* Bv        = (const float*)d_in[4];
    const float* Dvec      = (const float*)d_in[5];
    const float* W_mix     = (const float*)d_in[6];
    const float* b_mix     = (const float*)d_in[7];
    float* out = (float*)d_out;

    // workspace layout (floats): Amat | G | Kf | y  => 260 MB total
    float* Amat = (float*)d_ws;
    float* G    = Amat + (size_t)HCH * NMOD;
    float* Kf   = G    + (size_t)NMOD * 2 * NF;
    float* y    = Kf   + (size_t)HCH * 2 * NF;

    amat_kernel<<<(HCH * NMOD) / 256, 256, 0, stream>>>(C, Bv, Amat);
    g_table_kernel<<<(NMOD * NF) / 256, 256, 0, stream>>>(log_decay, phase, G);

    // Kf = Amat @ G : M=1024 -> 8 row-groups; Ncols=16384 -> 256 col-groups
    gemm_coop_kernel<<<dim3(8, 256), 256, 0, stream>>>(Amat, G, Kf, NMOD, 2 * NF);

    fftconv_kernel<<<BSZ * HCH, 256, 0, stream>>>(u, Kf, Dvec, y);

    // mix: 8 row-groups x 512 col-groups
    mix_gate_kernel<<<dim3(8, 512), 256, 0, stream>>>(W_mix, b_mix, y, out);
}